// GINRegressor_2327872274535
// MI455X (gfx1250) — compile-verified
//
#include <hip/hip_runtime.h>
#include <hip/hip_bf16.h>

typedef float v2f __attribute__((ext_vector_type(2)));
typedef float v8f __attribute__((ext_vector_type(8)));

#define FDIM 128
#define ROWS_PB 64   // rows of A per block in the GEMM (4 row-tiles of 16)

// ---------------------------------------------------------------------------
// CSR construction: degree count -> exclusive scan -> fill (atomic cursor)
// ---------------------------------------------------------------------------
__global__ void zero_int_kernel(int* __restrict__ p, int n) {
    int i = blockIdx.x * blockDim.x + threadIdx.x;
    if (i < n) p[i] = 0;
}

__global__ void count_deg_kernel(const int* __restrict__ dst, int* __restrict__ deg, int e) {
    int i = blockIdx.x * blockDim.x + threadIdx.x;
    if (i < e) atomicAdd(&deg[dst[i]], 1);
}

// Single-block chunked exclusive scan (n ~ 50k, trivial cost).
__global__ void excl_scan_kernel(const int* __restrict__ in, int* __restrict__ out, int n) {
    __shared__ int tmp[1024];
    __shared__ int carry_s;
    if (threadIdx.x == 0) carry_s = 0;
    __syncthreads();
    for (int base = 0; base < n; base += 1024) {
        int i = base + (int)threadIdx.x;
        int v = (i < n) ? in[i] : 0;
        tmp[threadIdx.x] = v;
        __syncthreads();
        for (int off = 1; off < 1024; off <<= 1) {   // Hillis-Steele inclusive
            int t = (threadIdx.x >= (unsigned)off) ? tmp[threadIdx.x - off] : 0;
            __syncthreads();
            tmp[threadIdx.x] += t;
            __syncthreads();
        }
        if (i < n) out[i] = carry_s + tmp[threadIdx.x] - v;   // exclusive
        __syncthreads();
        if (threadIdx.x == 1023) carry_s += tmp[1023];
        __syncthreads();
    }
    if (threadIdx.x == 0) out[n] = carry_s;                   // total = E
}

__global__ void fill_csr_kernel(const int* __restrict__ src, const int* __restrict__ dst,
                                const int* __restrict__ row_ptr, int* __restrict__ cursor,
                                int* __restrict__ col_idx, int e) {
    int i = blockIdx.x * blockDim.x + threadIdx.x;
    if (i < e) {
        int d = dst[i];
        int pos = row_ptr[d] + atomicAdd(&cursor[d], 1);
        col_idx[pos] = src[i];
    }
}

// ---------------------------------------------------------------------------
// agg[n][f] = (1+eps)*h[n][f] + sum_{j in N(n)} h[j][f]   (gather over CSR)
// one block (128 threads) per node; rows are L2-resident
// ---------------------------------------------------------------------------
__global__ void gin_aggregate_kernel(const float* __restrict__ h,
                                     const int* __restrict__ row_ptr,
                                     const int* __restrict__ col_idx,
                                     const float* __restrict__ eps,
                                     float* __restrict__ out, int n) {
    int node = blockIdx.x;
    if (node >= n) return;
    int f = threadIdx.x;  // 0..127
    int beg = row_ptr[node], end = row_ptr[node + 1];
    float s = 0.0f;
    for (int j = beg; j < end; ++j)
        s += h[(size_t)col_idx[j] * FDIM + f];
    out[(size_t)node * FDIM + f] = (1.0f + eps[0]) * h[(size_t)node * FDIM + f] + s;
}

// ---------------------------------------------------------------------------
// C[rows x 128] = act(A[rows x 128] @ W[128 x 128] + bias)
//
// Block = 256 threads = 8 waves; block owns ROWS_PB=64 rows (4 tiles of 16).
// Wave w owns output cols [16w, 16w+16).
//  * B column-tile (128x16 f32) is preloaded into 64 VGPRs per lane, read
//    straight from L2-resident W (read exactly once per block).
//  * The 64x128 A slab is staged into LDS once via the CDNA5 async-copy path
//    (global_load_async_to_lds_b128, ASYNCcnt), then broadcast to all 8
//    waves through ds_load — removes the 8x redundant global A reads.
//  * K=128 swept in steps of 4 with V_WMMA_F32_16X16X4_F32; 128 wmma/wave.
//
// f32 16x16x4 fragment layout (ISA 7.12.2):
//   A (16x4): lanes 0-15 row M=lane, VGPR0/1 = K0/K1; lanes 16-31 K2/K3
//   B (4x16): lanes 0-15 col N=lane, VGPR0/1 = K0/K1; lanes 16-31 K2/K3
//   C (16x16): VGPR i -> M=i (lanes 0-15) / M=i+8 (lanes 16-31), N = lane%16
// ---------------------------------------------------------------------------
__global__ __launch_bounds__(256)
void gin_gemm128_kernel(const float* __restrict__ A, const float* __restrict__ W,
                        const float* __restrict__ bias, float* __restrict__ C,
                        int nrows, int do_relu) {
    __shared__ float Alds[ROWS_PB * FDIM];  // 32 KB
    const int tid  = threadIdx.x;
    const int wave = tid >> 5;              // 0..7 -> N tile
    const int lane = tid & 31;
    const int hl   = lane & 15;             // 0..15
    const int koff = (lane >> 4) << 1;      // 0 or 2
    const int row0 = blockIdx.x * ROWS_PB;
    const int col0 = wave * 16;

    // ---- async-stage the 64x128 A slab into LDS (16B per lane per trip) ----
    // Low 32 bits of a flat LDS pointer are the wave-relative LDS byte
    // offset, usable as VDST.
    for (int idx = tid; idx < (ROWS_PB * FDIM) / 4; idx += 256) {
        int r  = idx >> 5;                  // 32 float4 chunks per row
        int cc = (idx & 31) << 2;
        int gr = min(row0 + r, nrows - 1);  // clamp tail rows
        const float* gp = A + (size_t)gr * FDIM + cc;
        unsigned lds_off = (unsigned)(uintptr_t)(&Alds[(r << 7) + cc]);
        asm volatile("global_load_async_to_lds_b128 %0, %1, off"
                     :: "v"(lds_off), "v"(gp)
                     : "memory");
    }
    asm volatile("s_wait_asynccnt 0x0" ::: "memory");
    __syncthreads();

    // ---- preload B column-tile into registers: bf[kk] covers K=4kk..4kk+3 ----
    v2f bf[32];
#pragma unroll
    for (int kk = 0; kk < 32; ++kk) {
        const int k = (kk << 2) + koff;
        bf[kk].x = W[(size_t)k * FDIM + col0 + hl];
        bf[kk].y = W[(size_t)(k + 1) * FDIM + col0 + hl];
    }

    const int   c  = col0 + hl;
    const float bv = bias[c];

    // ---- 4 row-tiles x 32 K-steps of f32 WMMA ----
    for (int rt = 0; rt < 4; ++rt) {
        const float* arow = &Alds[((rt << 4) + hl) << 7];  // (rt*16+hl)*128
        v8f acc = {};
#pragma unroll
        for (int kk = 0; kk < 32; ++kk) {
            v2f af;
            af.x = arow[(kk << 2) + koff];
            af.y = arow[(kk << 2) + koff + 1];
            acc = __builtin_amdgcn_wmma_f32_16x16x4_f32(
                /*neg_a=*/false, af, /*neg_b=*/false, bf[kk],
                /*c_mod=*/(short)0, acc, /*reuse_a=*/false, /*reuse_b=*/false);
        }
#pragma unroll
        for (int i = 0; i < 8; ++i) {
            int r = row0 + (rt << 4) + ((lane < 16) ? i : (i + 8));
            if (r < nrows) {
                float v = acc[i] + bv;
                if (do_relu) v = fmaxf(v, 0.0f);
                C[(size_t)r * FDIM + c] = v;
            }
        }
    }
}

// ---------------------------------------------------------------------------
// Fused mean-pool + final FC. batch is sorted, so graph g's nodes are the
// contiguous range found by binary search -> deterministic reduction.
// One block (128 threads) per graph.
// ---------------------------------------------------------------------------
__global__ void pool_fc_kernel(const float* __restrict__ h, const int* __restrict__ batch,
                               const float* __restrict__ fc_w, const float* __restrict__ fc_b,
                               float* __restrict__ out, int n, int g_total) {
    int g = blockIdx.x;
    if (g >= g_total) return;
    __shared__ int slo, shi;
    if (threadIdx.x == 0) {
        int lo = 0, hi = n;
        while (lo < hi) { int m = (lo + hi) >> 1; if (batch[m] <  g) lo = m + 1; else hi = m; }
        slo = lo;
        hi = n;
        while (lo < hi) { int m = (lo + hi) >> 1; if (batch[m] <= g) lo = m + 1; else hi = m; }
        shi = lo;
    }
    __syncthreads();
    const int f = threadIdx.x;  // 0..127
    float s = 0.0f;
    for (int nd = slo; nd < shi; ++nd) s += h[(size_t)nd * FDIM + f];
    float cnt = (float)(shi - slo);
    float pooled = s / fmaxf(cnt, 1.0f);

    __shared__ float red[FDIM];
    red[f] = pooled * fc_w[f];
    __syncthreads();
    for (int off = FDIM / 2; off > 0; off >>= 1) {
        if (f < off) red[f] += red[f + off];
        __syncthreads();
    }
    if (f == 0) out[g] = red[0] + fc_b[0];
}

// ---------------------------------------------------------------------------
// Host orchestration
// ---------------------------------------------------------------------------
extern "C" void kernel_launch(void* const* d_in, const int* in_sizes, int n_in,
                              void* d_out, int out_size, void* d_ws, size_t ws_size,
                              hipStream_t stream) {
    const float* x          = (const float*)d_in[0];
    const int*   edge_index = (const int*)d_in[1];
    const int*   batch      = (const int*)d_in[2];
    const float* W1[3] = { (const float*)d_in[3],  (const float*)d_in[8],  (const float*)d_in[13] };
    const float* b1[3] = { (const float*)d_in[4],  (const float*)d_in[9],  (const float*)d_in[14] };
    const float* W2[3] = { (const float*)d_in[5],  (const float*)d_in[10], (const float*)d_in[15] };
    const float* b2[3] = { (const float*)d_in[6],  (const float*)d_in[11], (const float*)d_in[16] };
    const float* eps[3]= { (const float*)d_in[7],  (const float*)d_in[12], (const float*)d_in[17] };
    const float* fc_w  = (const float*)d_in[18];
    const float* fc_b  = (const float*)d_in[19];

    const int N = in_sizes[2];
    const int E = in_sizes[1] / 2;
    const int G = out_size;
    const int* src = edge_index;      // row 0
    const int* dst = edge_index + E;  // row 1

    // workspace carve-up (256B aligned)
    char* ws = (char*)d_ws;
    size_t off = 0;
    auto carve = [&](size_t bytes) -> void* {
        void* p = ws + off;
        off = (off + bytes + 255) & ~(size_t)255;
        return p;
    };
    int*   cursor  = (int*)carve((size_t)N * sizeof(int));
    int*   row_ptr = (int*)carve((size_t)(N + 1) * sizeof(int));
    int*   col_idx = (int*)carve((size_t)E * sizeof(int));
    float* Abuf    = (float*)carve((size_t)N * FDIM * sizeof(float));
    float* Tbuf    = (float*)carve((size_t)N * FDIM * sizeof(float));
    float* Hbuf    = (float*)carve((size_t)N * FDIM * sizeof(float));
    (void)ws_size; (void)n_in;

    const int TPB = 256;
    // --- build CSR (dst -> list of src) ---
    zero_int_kernel<<<(N + TPB - 1) / TPB, TPB, 0, stream>>>(cursor, N);
    count_deg_kernel<<<(E + TPB - 1) / TPB, TPB, 0, stream>>>(dst, cursor, E);
    excl_scan_kernel<<<1, 1024, 0, stream>>>(cursor, row_ptr, N);
    zero_int_kernel<<<(N + TPB - 1) / TPB, TPB, 0, stream>>>(cursor, N);
    fill_csr_kernel<<<(E + TPB - 1) / TPB, TPB, 0, stream>>>(src, dst, row_ptr, cursor, col_idx, E);

    const int gemm_blocks = (N + ROWS_PB - 1) / ROWS_PB;
    const float* h_in = x;
    for (int l = 0; l < 3; ++l) {
        gin_aggregate_kernel<<<N, FDIM, 0, stream>>>(h_in, row_ptr, col_idx, eps[l], Abuf, N);
        gin_gemm128_kernel<<<gemm_blocks, 256, 0, stream>>>(Abuf, W1[l], b1[l], Tbuf, N, /*relu=*/1);
        gin_gemm128_kernel<<<gemm_blocks, 256, 0, stream>>>(Tbuf, W2[l], b2[l], Hbuf, N, /*relu=*/(l < 2) ? 1 : 0);
        h_in = Hbuf;
    }

    pool_fc_kernel<<<G, FDIM, 0, stream>>>(Hbuf, batch, fc_w, fc_b, (float*)d_out, N, G);
}